// RPN_37486474559704
// MI455X (gfx1250) — compile-verified
//
#include <hip/hip_runtime.h>
#include <hip/hip_bf16.h>
#include <stdint.h>

// ---------------- problem constants (match reference) ----------------
#define N0 3145728   // 16*3*256*256
#define N1 786432    // 16*3*128*128
#define N2 196608    // 16*3*64*64
#define NT (N0 + N1 + N2)   // 4,128,768 total anchors/scores
#define TOPK 2000
#define KPAD 2048
#define CAP  4096
#define PROP_THR 0.7f
#define NMS_THR  0.7f

// ---------------- workspace layout (bytes) ----------------
#define OFF_H1    0         // 4096 u32 hist pass1
#define OFF_H2    16384     // 4096 u32 hist pass2
#define OFF_SEL   32768     // 16 u32: [0]=b1, [1]=C1, [2]=T threshold key
#define OFF_CNT   32832     // 1 u32 compact counter (padded to 64B)
#define OFF_CAND  32896     // 4096 x uint2 (key, idx)
#define OFF_TKEY  65664     // 2048 u32 sorted top keys (0 = invalid)
#define OFF_TIDX  73856     // 2048 u32 sorted top global indices
#define OFF_TB    82048     // 2048 x float4 decoded top boxes (x,y,w,h)
#define OFF_TCOR  114816    // 2048 x float4 corners (x1,y1,x2,y2)
#define OFF_TAREA 147584    // 2048 f32 clipped areas
#define OFF_MASK  155776    // 2048*64 u32 suppression bitmask rows
#define WS_END    680064
#define WS_WORDS  (WS_END / 4)

typedef __attribute__((ext_vector_type(2))) float v2f;
typedef __attribute__((ext_vector_type(8))) float v8f;

// ---------------- 0: zero the workspace (hists/counter/mask must be 0) ----
__global__ void k_zero(uint32_t* ws) {
  int i = blockIdx.x * blockDim.x + threadIdx.x;
  if (i < WS_WORDS) ws[i] = 0u;
}

// ---------------- 1/2: radix-select histograms over score bit patterns ----
// Scores are cls values flattened in concat(cls0,cls1,cls2) order. Positive-
// float bits sort like the floats themselves; key = bits if score>0.7 else 0.
__global__ __launch_bounds__(256) void k_hist(const float* __restrict__ c0,
                                              const float* __restrict__ c1,
                                              const float* __restrict__ c2,
                                              uint32_t* __restrict__ hist,
                                              const uint32_t* __restrict__ sel,
                                              int mode) {
  __shared__ uint32_t h[4096];
  for (int i = threadIdx.x; i < 4096; i += blockDim.x) h[i] = 0u;
  __syncthreads();
  uint32_t b1 = sel[0];
  int stride = gridDim.x * blockDim.x;
  for (int g = blockIdx.x * blockDim.x + threadIdx.x; g < NT; g += stride) {
    float s = (g < N0) ? c0[g] : ((g < N0 + N1) ? c1[g - N0] : c2[g - N0 - N1]);
    if (s > PROP_THR) {
      uint32_t key = __float_as_uint(s);
      if (mode == 0)              atomicAdd(&h[key >> 20], 1u);
      else if ((key >> 20) == b1) atomicAdd(&h[(key >> 8) & 0xFFFu], 1u);
    }
  }
  __syncthreads();
  for (int i = threadIdx.x; i < 4096; i += blockDim.x)
    if (h[i]) atomicAdd(&hist[i], h[i]);
}

// ---------------- scan: pick crossing bin (single thread; tiny) -----------
__global__ void k_scan(const uint32_t* __restrict__ hist, uint32_t* sel, int mode) {
  if (threadIdx.x != 0 || blockIdx.x != 0) return;
  if (mode == 0) {
    uint32_t acc = 0;
    for (int i = 4095; i >= 0; --i) {
      uint32_t c = hist[i];
      if (acc + c >= TOPK) { sel[0] = (uint32_t)i; sel[1] = acc; return; }
      acc += c;
    }
    sel[0] = 0xFFFFFFFFu;  // fewer than K valid: admit every valid key
    sel[1] = acc;
    sel[2] = 1u;
  } else {
    if (sel[0] == 0xFFFFFFFFu) return;  // threshold already set to 1
    uint32_t acc = sel[1];
    for (int i = 4095; i >= 0; --i) {
      acc += hist[i];
      if (acc >= TOPK) { sel[2] = (sel[0] << 20) | ((uint32_t)i << 8); return; }
    }
    sel[2] = (sel[0] << 20);  // defensive fallback
  }
}

// ---------------- 3: compact candidates with key >= T ---------------------
__global__ __launch_bounds__(256) void k_compact(const float* __restrict__ c0,
                                                 const float* __restrict__ c1,
                                                 const float* __restrict__ c2,
                                                 const uint32_t* __restrict__ sel,
                                                 uint32_t* __restrict__ counter,
                                                 uint2* __restrict__ cand) {
  uint32_t T = sel[2];
  int stride = gridDim.x * blockDim.x;
  for (int g = blockIdx.x * blockDim.x + threadIdx.x; g < NT; g += stride) {
    float s = (g < N0) ? c0[g] : ((g < N0 + N1) ? c1[g - N0] : c2[g - N0 - N1]);
    if (s > PROP_THR) {
      uint32_t key = __float_as_uint(s);
      if (key >= T) {
        uint32_t p = atomicAdd(counter, 1u);
        if (p < CAP) cand[p] = make_uint2(key, (uint32_t)g);
      }
    }
  }
}

// ---------------- 4: exact top-K via bitonic sort of <=4096 cands ---------
// Composite 64-bit key (scorebits<<32)|~idx sorted descending reproduces
// jax.lax.top_k ordering (score desc, index asc on ties). Sentinel = 0.
__global__ __launch_bounds__(1024) void k_sort(const uint32_t* __restrict__ counter,
                                               const uint2* __restrict__ cand,
                                               uint32_t* __restrict__ tkey,
                                               uint32_t* __restrict__ tidx) {
  __shared__ unsigned long long a[4096];   // 32 KB of 320 KB LDS
  uint32_t n = *counter; if (n > CAP) n = CAP;
  for (int i = threadIdx.x; i < 4096; i += 1024) {
    if (i < (int)n) {
      uint2 c = cand[i];
      a[i] = (((unsigned long long)c.x) << 32) | (uint32_t)(~c.y);
    } else a[i] = 0ull;
  }
  __syncthreads();
  for (int k = 2; k <= 4096; k <<= 1) {
    for (int j = k >> 1; j > 0; j >>= 1) {
      for (int i = threadIdx.x; i < 4096; i += 1024) {
        int p = i ^ j;
        if (p > i) {
          bool desc = ((i & k) == 0);
          unsigned long long x = a[i], y = a[p];
          if (desc ? (x < y) : (x > y)) { a[i] = y; a[p] = x; }
        }
      }
      __syncthreads();
    }
  }
  for (int s = threadIdx.x; s < KPAD; s += 1024) {
    unsigned long long v = a[s];
    tkey[s] = (uint32_t)(v >> 32);
    tidx[s] = ~(uint32_t)(v & 0xFFFFFFFFull);
  }
}

// ---------------- 5: decode boxes only for the 2000 winners ---------------
// Also precompute corners and clipped areas once per box so the O(K^2) IoU
// stage does no redundant per-tile corner math.
__global__ __launch_bounds__(256) void k_decode(const uint32_t* __restrict__ tkey,
                                                const uint32_t* __restrict__ tidx,
                                                const float* __restrict__ bb0, const float* __restrict__ an0,
                                                const float* __restrict__ bb1, const float* __restrict__ an1,
                                                const float* __restrict__ bb2, const float* __restrict__ an2,
                                                float4* __restrict__ tb,
                                                float4* __restrict__ tcor,
                                                float* __restrict__ tarea) {
  int s = blockIdx.x * blockDim.x + threadIdx.x;
  if (s >= KPAD) return;
  if (s >= TOPK || tkey[s] == 0u) {
    tb[s]    = make_float4(0.f, 0.f, 0.f, 0.f);
    tcor[s]  = make_float4(0.f, 0.f, 0.f, 0.f);
    tarea[s] = 0.f;
    return;
  }
  uint32_t g = tidx[s];
  const float* bb; const float* an; int H; uint32_t l;
  if (g < N0)            { bb = bb0; an = an0; H = 256; l = g; }
  else if (g < N0 + N1)  { bb = bb1; an = an1; H = 128; l = g - N0; }
  else                   { bb = bb2; an = an2; H = 64;  l = g - N0 - N1; }
  int HH = H * H;
  int b  = (int)(l / (uint32_t)(3 * HH));
  int r  = (int)(l % (uint32_t)(3 * HH));
  int kk = r / HH;
  int rr = r % HH;                       // y*H + x
  size_t base = (size_t)(b * 12 + kk * 4) * HH + rr;
  float dx = bb[base], dy = bb[base + HH], dw = bb[base + 2 * (size_t)HH], dh = bb[base + 3 * (size_t)HH];
  const float* a4 = an + ((size_t)kk * HH + rr) * 4;
  float ax = a4[0], ay = a4[1], aw = a4[2], ah = a4[3];
  float x = ax + dx * aw, y = ay + dy * ah;
  float w = aw * expf(dw), h = ah * expf(dh);
  tb[s]    = make_float4(x, y, w, h);
  tcor[s]  = make_float4(x - 0.5f * w, y - 0.5f * h, x + 0.5f * w, y + 0.5f * h);
  tarea[s] = fmaxf(w, 0.f) * fmaxf(h, 0.f);
}

// ---------------- 6: 16x16 IoU tiles; WMMA computes area_i + area_j -------
// One wave per tile. A = [area_m, 1, 0, 0] (16x4), B = [[1],[area_n],[0],[0]]
// (4x16) => D[m][n] = area_m + area_n via V_WMMA_F32_16X16X4_F32. C/D layout:
// VGPR r -> M = r (+8 for lanes 16..31), N = lane & 15. The IoU compare is
// done division-free: inter/max(uni,eps) > thr  <=>  inter > thr*max(uni,eps).
__global__ __launch_bounds__(32) void k_iou(const float4* __restrict__ tcor,
                                            const float* __restrict__ tarea,
                                            uint32_t* __restrict__ mask) {
  int i0 = blockIdx.y * 16, j0 = blockIdx.x * 16;
  int lane = threadIdx.x;
  __shared__ float4 rc[16], cc[16];
  __shared__ float  ra[16], ca[16];
  if (lane < 16) { rc[lane] = tcor[i0 + lane];        ra[lane] = tarea[i0 + lane]; }
  else           { cc[lane - 16] = tcor[j0 + lane - 16]; ca[lane - 16] = tarea[j0 + lane - 16]; }
  __syncthreads();

  int n = lane & 15;
  float4 c  = cc[n];          // (x1,y1,x2,y2) of column box
  float carea = ca[n];
  float rarea = ra[n];        // lane m holds row-m area for the A operand

  bool lo = lane < 16;
  v2f A; A[0] = lo ? rarea : 0.f; A[1] = lo ? 1.f : 0.f;   // K=0,1 (lanes<16); K=2,3 zero
  v2f B; B[0] = lo ? 1.f : 0.f;   B[1] = lo ? carea : 0.f; // rows K=0..3
  v8f C = {};
  C = __builtin_amdgcn_wmma_f32_16x16x4_f32(false, A, false, B, (short)0, C, false, false);

  int half8 = (lane >> 4) * 8;
  uint32_t bits[8];
#pragma unroll
  for (int r = 0; r < 8; ++r) {
    int m = r + half8;
    float4 rm = rc[m];        // (x1,y1,x2,y2) of row box
    float iw = fminf(rm.z, c.z) - fmaxf(rm.x, c.x);
    float ih = fminf(rm.w, c.w) - fmaxf(rm.y, c.y);
    float inter = fmaxf(iw, 0.f) * fmaxf(ih, 0.f);
    float uni = fmaxf(C[r] - inter, 1e-6f);    // (area_m + area_n) from WMMA
    int gi = i0 + m, gj = j0 + n;
    bool sup = (inter > NMS_THR * uni) && (gj > gi);   // division-free IoU test
    bits[r] = __builtin_amdgcn_ballot_w32(sup);        // wave32 ballot
  }
  if (lane == 0) {
    int word = j0 >> 5, sh = j0 & 31;                  // sh = 0 or 16
    for (int r = 0; r < 8; ++r) {
      atomicOr(&mask[(size_t)(i0 + r) * 64 + word], (bits[r] & 0xFFFFu) << sh);
      atomicOr(&mask[(size_t)(i0 + r + 8) * 64 + word], (bits[r] >> 16) << sh);
    }
  }
}

// ---------------- 7: sequential greedy NMS + final outputs ----------------
__global__ __launch_bounds__(64) void k_nms(const uint32_t* __restrict__ tkey,
                                            const float4* __restrict__ tb,
                                            const uint32_t* __restrict__ mask,
                                            float* __restrict__ out) {
  __shared__ uint32_t rem[64];   // 2048-bit removed set
  int t = threadIdx.x;
  rem[t] = 0u;
  __syncthreads();
  for (int i = 0; i < TOPK; ++i) {
    uint32_t w = rem[i >> 5];
    bool alive = (((w >> (i & 31)) & 1u) == 0u) && (tkey[i] != 0u);
    __syncthreads();
    if (alive) rem[t] |= mask[(size_t)i * 64 + t];
    __syncthreads();
  }
  for (int s = t; s < TOPK; s += 64) {
    bool keep = (tkey[s] != 0u) && (((rem[s >> 5] >> (s & 31)) & 1u) == 0u);
    float f = keep ? 1.f : 0.f;
    float4 b = tb[s];
    out[s * 4 + 0] = b.x * f;
    out[s * 4 + 1] = b.y * f;
    out[s * 4 + 2] = b.z * f;
    out[s * 4 + 3] = b.w * f;
    out[4 * TOPK + s] = f;     // keep mask, concatenated after boxes
  }
}

// ---------------- host-side launcher --------------------------------------
extern "C" void kernel_launch(void* const* d_in, const int* in_sizes, int n_in,
                              void* d_out, int out_size, void* d_ws, size_t ws_size,
                              hipStream_t stream) {
  // setup_inputs dict order: cls0, bbox0, anc0, cls1, bbox1, anc1, cls2, bbox2, anc2
  const float* cls0 = (const float*)d_in[0];
  const float* bb0  = (const float*)d_in[1];
  const float* an0  = (const float*)d_in[2];
  const float* cls1 = (const float*)d_in[3];
  const float* bb1  = (const float*)d_in[4];
  const float* an1  = (const float*)d_in[5];
  const float* cls2 = (const float*)d_in[6];
  const float* bb2  = (const float*)d_in[7];
  const float* an2  = (const float*)d_in[8];
  (void)in_sizes; (void)n_in; (void)out_size; (void)ws_size;

  uint8_t*  w     = (uint8_t*)d_ws;
  uint32_t* h1    = (uint32_t*)(w + OFF_H1);
  uint32_t* h2    = (uint32_t*)(w + OFF_H2);
  uint32_t* sel   = (uint32_t*)(w + OFF_SEL);
  uint32_t* cnt   = (uint32_t*)(w + OFF_CNT);
  uint2*    cand  = (uint2*)   (w + OFF_CAND);
  uint32_t* tkey  = (uint32_t*)(w + OFF_TKEY);
  uint32_t* tidx  = (uint32_t*)(w + OFF_TIDX);
  float4*   tb    = (float4*)  (w + OFF_TB);
  float4*   tcor  = (float4*)  (w + OFF_TCOR);
  float*    tarea = (float*)   (w + OFF_TAREA);
  uint32_t* mask  = (uint32_t*)(w + OFF_MASK);

  k_zero<<<(WS_WORDS + 255) / 256, 256, 0, stream>>>((uint32_t*)d_ws);

  k_hist<<<2048, 256, 0, stream>>>(cls0, cls1, cls2, h1, sel, 0);
  k_scan<<<1, 1, 0, stream>>>(h1, sel, 0);
  k_hist<<<2048, 256, 0, stream>>>(cls0, cls1, cls2, h2, sel, 1);
  k_scan<<<1, 1, 0, stream>>>(h2, sel, 1);
  k_compact<<<2048, 256, 0, stream>>>(cls0, cls1, cls2, sel, cnt, cand);
  k_sort<<<1, 1024, 0, stream>>>(cnt, cand, tkey, tidx);
  k_decode<<<KPAD / 256, 256, 0, stream>>>(tkey, tidx, bb0, an0, bb1, an1, bb2, an2,
                                           tb, tcor, tarea);
  k_iou<<<dim3(KPAD / 16, KPAD / 16), 32, 0, stream>>>(tcor, tarea, mask);
  k_nms<<<1, 64, 0, stream>>>(tkey, tb, mask, (float*)d_out);
}